// PositionWeightsAttacher_49409303773882
// MI455X (gfx1250) — compile-verified
//
#include <hip/hip_runtime.h>
#include <hip/hip_bf16.h>
#include <stdint.h>

// ---------------------------------------------------------------------------
// PositionWeightsAttacher for MI455X (gfx1250, wave32)
//   out[e] = pw[e - seg_start(e)]  for two features, concatenated.
// Pure bandwidth problem (~170MB traffic -> ~7.3us @ 23.3TB/s). Strategy:
//   scan lengths -> per-sample offsets, then one wave32 per sample writes
//   pw[0..len) with coalesced 128B store bursts and non-temporal hints.
//   pw_f1 is staged into LDS via the CDNA5 Tensor Data Mover.
// ---------------------------------------------------------------------------

static constexpr int BATCH      = 524288;
static constexpr int MAXLEN_F1  = 128;
static constexpr int MAXLEN_F2  = 20;
static constexpr long long TOTAL_F1 = (long long)BATCH * (MAXLEN_F1 / 2);
static constexpr int SCAN_BLOCK = 256;
static constexpr int SCAN_ITEMS = 16;
static constexpr int SCAN_CHUNK = SCAN_BLOCK * SCAN_ITEMS;   // 4096
static constexpr int NSCAN      = BATCH / SCAN_CHUNK;        // 128

#if defined(__gfx1250__) && __has_builtin(__builtin_amdgcn_tensor_load_to_lds) && __has_builtin(__builtin_amdgcn_s_wait_tensorcnt)
#define PWA_USE_TDM 1
#else
#define PWA_USE_TDM 0
#endif

typedef unsigned int uint32;
typedef uint32 v4u __attribute__((ext_vector_type(4)));
typedef int    v4i __attribute__((ext_vector_type(4)));
typedef int    v8i __attribute__((ext_vector_type(8)));

// --- K1: per-block exclusive scan of lengths (both features in one launch) --
__global__ void pwa_scan_local(const int* __restrict__ len1, const int* __restrict__ len2,
                               int* __restrict__ offs1, int* __restrict__ offs2,
                               int* __restrict__ bsum1, int* __restrict__ bsum2) {
  __shared__ int sdata[SCAN_BLOCK];
  const bool second = blockIdx.x >= (unsigned)NSCAN;
  const int  blk    = second ? (int)blockIdx.x - NSCAN : (int)blockIdx.x;
  const int* __restrict__ lens = second ? len2 : len1;
  int* __restrict__ offs = second ? offs2 : offs1;
  int* __restrict__ bsum = second ? bsum2 : bsum1;

  const int tid  = threadIdx.x;
  const int base = blk * SCAN_CHUNK + tid * SCAN_ITEMS;

  int v[SCAN_ITEMS];
#pragma unroll
  for (int i = 0; i < SCAN_ITEMS; ++i) v[i] = lens[base + i];
  int sum = 0;
#pragma unroll
  for (int i = 0; i < SCAN_ITEMS; ++i) { int t = v[i]; v[i] = sum; sum += t; }

  sdata[tid] = sum;
  __syncthreads();
  for (int off = 1; off < SCAN_BLOCK; off <<= 1) {
    int t = (tid >= off) ? sdata[tid - off] : 0;
    __syncthreads();
    sdata[tid] += t;
    __syncthreads();
  }
  const int inclusive = sdata[tid];
  const int exclBase  = inclusive - sum;
#pragma unroll
  for (int i = 0; i < SCAN_ITEMS; ++i) offs[base + i] = exclBase + v[i];
  if (tid == SCAN_BLOCK - 1) bsum[blk] = inclusive;
}

// --- K2: exclusive scan of the 128 block sums (one tiny block per feature) --
__global__ void pwa_scan_bsums(int* __restrict__ bsum1, int* __restrict__ bsum2) {
  __shared__ int sdata[NSCAN];
  int* __restrict__ b = blockIdx.x ? bsum2 : bsum1;
  const int tid = threadIdx.x;
  const int v = b[tid];
  sdata[tid] = v;
  __syncthreads();
  for (int off = 1; off < NSCAN; off <<= 1) {
    int t = (tid >= off) ? sdata[tid - off] : 0;
    __syncthreads();
    sdata[tid] += t;
    __syncthreads();
  }
  b[tid] = sdata[tid] - v;   // exclusive block offset
}

// --- K3: feature-1 fill. One wave32 per sample, 4 coalesced store rounds. ---
__global__ void pwa_fill_f1(const float* __restrict__ pw, const int* __restrict__ lens,
                            const int* __restrict__ offs, const int* __restrict__ bsum,
                            float* __restrict__ out) {
  __shared__ float spw[MAXLEN_F1];
  const int tid = threadIdx.x;

#if PWA_USE_TDM
  // Stage the 512B weight table into LDS with one Tensor Data Mover DMA.
  // D# group0: count=1 | lds_addr | global_addr | type=2 ("image").
  // D# group1: data_size=4B, tensor_dim0=128, tensor_dim1=1, tile=128x1,
  //            dim0_stride=128. Groups 2/3 zero (2D tensor).
  // amdgpu-toolchain (clang-23) 6-arg builtin form.
  const int wv = __builtin_amdgcn_readfirstlane(tid >> 5);
  if (wv == 0) {
    const unsigned long long ga = (unsigned long long)(uintptr_t)pw;
    const uint32 ldsOff = (uint32)(uintptr_t)(void*)&spw[0];  // low bits = LDS byte offset
    v4u g0;
    g0[0] = 1u;                                     // count=1, user descriptor
    g0[1] = ldsOff;                                 // lds_addr
    g0[2] = (uint32)ga;                             // global_addr[31:0]
    g0[3] = (uint32)((ga >> 32) & 0x01FFFFFFu) | (2u << 30);  // addr[56:32] | type=2
    v8i g1;
    g1[0] = (int)(2u << 16);          // workgroup_mask=0, data_size=2 (4 bytes)
    g1[1] = (int)(128u << 16);        // tensor_dim0[15:0] at bit 48
    g1[2] = (int)(1u << 16);          // tensor_dim0 hi=0 | tensor_dim1=1 at bit 80
    g1[3] = (int)(128u << 16);        // tensor_dim1 hi=0 | tile_dim0=128
    g1[4] = 1;                        // tile_dim1=1, tile_dim2=0
    g1[5] = 128;                      // tensor_dim0_stride lo
    g1[6] = 0;                        // stride0 hi | stride1 lo
    g1[7] = 0;
    v4i z4 = {0, 0, 0, 0};
    v8i z8 = {0, 0, 0, 0, 0, 0, 0, 0};
    __builtin_amdgcn_tensor_load_to_lds(g0, g1, z4, z4, z8, 0);
    __builtin_amdgcn_s_wait_tensorcnt(0);
  }
#else
  if (tid < MAXLEN_F1) spw[tid] = pw[tid];
#endif
  __syncthreads();

  const int lane = tid & 31;
  const float r0 = spw[lane];
  const float r1 = spw[lane + 32];
  const float r2 = spw[lane + 64];
  const float r3 = spw[lane + 96];

  const int gw = (int)((blockIdx.x * blockDim.x + tid) >> 5);
  const int nw = (int)((gridDim.x * blockDim.x) >> 5);
  for (int s = gw; s < BATCH; s += nw) {
    const int len = lens[s];
    const long long o = (long long)offs[s] + bsum[s >> 12];
    float* __restrict__ p = out + o;
    if (lane      < len) __builtin_nontemporal_store(r0, p + lane);
    if (lane + 32 < len) __builtin_nontemporal_store(r1, p + lane + 32);
    if (lane + 64 < len) __builtin_nontemporal_store(r2, p + lane + 64);
    if (lane + 96 < len) __builtin_nontemporal_store(r3, p + lane + 96);
  }
}

// --- K4: feature-2 fill. One wave32 per sample, single store round. --------
__global__ void pwa_fill_f2(const float* __restrict__ pw, const int* __restrict__ lens,
                            const int* __restrict__ offs, const int* __restrict__ bsum,
                            float* __restrict__ out) {
  __shared__ float spw[MAXLEN_F2];
  const int tid = threadIdx.x;
  if (tid < MAXLEN_F2) spw[tid] = pw[tid];
  __syncthreads();

  const int lane = tid & 31;
  const float r = spw[lane < MAXLEN_F2 ? lane : 0];

  const int gw = (int)((blockIdx.x * blockDim.x + tid) >> 5);
  const int nw = (int)((gridDim.x * blockDim.x) >> 5);
  for (int s = gw; s < BATCH; s += nw) {
    const int len = lens[s];                       // <= 20 < 32
    const long long o = (long long)offs[s] + bsum[s >> 12];
    if (lane < len) __builtin_nontemporal_store(r, out + o + lane);
  }
}

extern "C" void kernel_launch(void* const* d_in, const int* in_sizes, int n_in,
                              void* d_out, int out_size, void* d_ws, size_t ws_size,
                              hipStream_t stream) {
  (void)in_sizes; (void)n_in; (void)out_size; (void)ws_size;
  const float* pw_f1 = (const float*)d_in[0];
  const float* pw_f2 = (const float*)d_in[1];
  const int*   len1  = (const int*)d_in[2];
  const int*   len2  = (const int*)d_in[3];
  float* out = (float*)d_out;

  // Workspace: offs1[B] | offs2[B] | bsum1[128] | bsum2[128]  (~4.2 MB)
  int* offs1 = (int*)d_ws;
  int* offs2 = offs1 + BATCH;
  int* bsum1 = offs2 + BATCH;
  int* bsum2 = bsum1 + NSCAN;

  pwa_scan_local<<<2 * NSCAN, SCAN_BLOCK, 0, stream>>>(len1, len2, offs1, offs2, bsum1, bsum2);
  pwa_scan_bsums<<<2, NSCAN, 0, stream>>>(bsum1, bsum2);
  pwa_fill_f1<<<1024, 256, 0, stream>>>(pw_f1, len1, offs1, bsum1, out);
  pwa_fill_f2<<<1024, 256, 0, stream>>>(pw_f2, len2, offs2, bsum2, out + TOTAL_F1);
}